// NKAD_Base_14654428414244
// MI455X (gfx1250) — compile-verified
//
#include <hip/hip_runtime.h>
#include <hip/hip_bf16.h>
#include <math.h>

// ---------------------------------------------------------------------------
// GAD-NR-style forward for MI455X (gfx1250).
// fp32 end-to-end; matrix work on V_WMMA_F32_16X16X4_F32 (wave32).
// Graph segment-ops are gather-based via CSR (no float atomics).
// generated_cov [N,64,64] is staged in the s_ region of d_out (written last).
// ---------------------------------------------------------------------------

typedef float v2f __attribute__((ext_vector_type(2)));
typedef float v8f __attribute__((ext_vector_type(8)));

#define HIDN   64
#define INDIM  128
#define KSAMP  10
#define MAXNEI 64

__device__ __forceinline__ v8f wmma_f32(v2f a, v2f b, v8f c) {
  // D = A(16x4 f32) * B(4x16 f32) + C(16x16 f32)
  return __builtin_amdgcn_wmma_f32_16x16x4_f32(false, a, false, b, (short)0, c,
                                               false, false);
}

// ------------------------------ small utils -------------------------------

__global__ void k_zero_i32(int* p, int n) {
  int i = blockIdx.x * blockDim.x + threadIdx.x;
  if (i < n) p[i] = 0;
}
__global__ void k_zero_f32(float* p, int n) {
  int i = blockIdx.x * blockDim.x + threadIdx.x;
  if (i < n) p[i] = 0.f;
}

// ------------------------------ CSR build ---------------------------------

__global__ void k_count_deg(const int* __restrict__ dstp, int* cnt, int E) {
  int e = blockIdx.x * blockDim.x + threadIdx.x;
  if (e < E) atomicAdd(&cnt[dstp[e]], 1);
}

__global__ void k_prefix(const int* __restrict__ cnt, int* rowptr, int n) {
  if (blockIdx.x == 0 && threadIdx.x == 0) {
    int acc = 0;
    rowptr[0] = 0;
    for (int i = 0; i < n; ++i) { acc += cnt[i]; rowptr[i + 1] = acc; }
  }
}

__global__ void k_fill_csr(const int* __restrict__ dstp,
                           const int* __restrict__ rowptr, int* fill, int* csr,
                           int E) {
  int e = blockIdx.x * blockDim.x + threadIdx.x;
  if (e >= E) return;
  int d = dstp[e];
  int pos = atomicAdd(&fill[d], 1);
  csr[rowptr[d] + pos] = e;
}

// stable order within a dst-group == ascending edge id (matches jnp.argsort)
__global__ void k_sort_csr(const int* __restrict__ rowptr, int* csr, int n) {
  int i = blockIdx.x * blockDim.x + threadIdx.x;
  if (i >= n) return;
  int beg = rowptr[i], end = rowptr[i + 1];
  for (int a = beg + 1; a < end; ++a) {
    int v = csr[a];
    int b = a - 1;
    while (b >= beg && csr[b] > v) { csr[b + 1] = csr[b]; --b; }
    csr[b + 1] = v;
  }
}

__global__ void k_dinv(const int* __restrict__ cnt, float* dsl, float* dns,
                       int n) {
  int i = blockIdx.x * blockDim.x + threadIdx.x;
  if (i >= n) return;
  float c = (float)cnt[i];
  dsl[i] = 1.f / sqrtf(c + 1.f);
  dns[i] = (c > 0.f) ? 1.f / sqrtf(c) : 0.f;
}

// ------------------------- WMMA GEMM  C = act(A*W+b) -----------------------
// A:[M,KA]  W:[KA,NW]  bias:[NW] (nullable)  out:[M,NW]
// grid.x = M/16 ; block = 32*(NW/16) threads ; wave w owns columns [16w,16w+16)
// MODE: 0 none, 1 relu, 2 prelu(0.25), 3 clip[1e-6,10]
template <int KA, int NW, int MODE>
__global__ __launch_bounds__(32 * (NW / 16)) void k_gemm(
    const float* __restrict__ A, const float* __restrict__ W,
    const float* __restrict__ bias, float* __restrict__ out, int M) {
  __shared__ float As[16][KA + 1];
  const int tid = threadIdx.x;
  const int row0 = blockIdx.x * 16;
  for (int i = tid; i < 16 * KA; i += 32 * (NW / 16)) {
    int r = i / KA, c = i % KA;
    As[r][c] = A[(size_t)(row0 + r) * KA + c];
  }
  __syncthreads();
  const int wave = tid >> 5;
  const int lane = tid & 31;
  const int half = lane >> 4;   // 0: K=0,1 / M rows r..  1: K=2,3 / M rows r+8
  const int l = lane & 15;
  const int col = wave * 16 + l;
  v8f acc = {};
  for (int k0 = 0; k0 < KA; k0 += 4) {
    int ka = k0 + 2 * half;
    v2f a; a.x = As[l][ka];               a.y = As[l][ka + 1];
    v2f b; b.x = W[(size_t)ka * NW + col]; b.y = W[(size_t)(ka + 1) * NW + col];
    acc = wmma_f32(a, b, acc);
  }
  float bc = bias ? bias[col] : 0.f;
  for (int r = 0; r < 8; ++r) {
    int m = r + 8 * half;                 // ISA 16x16 f32 C/D layout
    float v = acc[r] + bc;
    if (MODE == 1) v = fmaxf(v, 0.f);
    else if (MODE == 2) v = (v > 0.f) ? v : 0.25f * v;
    else if (MODE == 3) v = fminf(fmaxf(v, 1e-6f), 10.f);
    out[(size_t)(row0 + m) * NW + col] = v;
  }
}

// ---------------- s_ = H * H^T  (H:[M,128]) -> out:[M,M] -------------------
// grid(M/16, ceil((M/16)/4)), block 128 (4 waves). NT stores (400MB streamed).
__global__ __launch_bounds__(128) void k_gemm_sim(const float* __restrict__ H,
                                                  float* __restrict__ out,
                                                  int M) {
  __shared__ float As[16][129];
  const int tid = threadIdx.x;
  const int row0 = blockIdx.x * 16;
  for (int i = tid; i < 16 * 128; i += 128) {
    int r = i >> 7, c = i & 127;
    As[r][c] = H[(size_t)(row0 + r) * 128 + c];
  }
  __syncthreads();
  const int wave = tid >> 5, lane = tid & 31, half = lane >> 4, l = lane & 15;
  const int ntile = blockIdx.y * 4 + wave;
  if (ntile * 16 >= M) return;          // uniform per-wave: EXEC stays all-1s
  const int coln = ntile * 16 + l;
  const float* Bp = H + (size_t)coln * 128;  // B[k][n] = H[n][k]
  v8f acc = {};
  for (int k0 = 0; k0 < 128; k0 += 4) {
    int ka = k0 + 2 * half;
    v2f a; a.x = As[l][ka]; a.y = As[l][ka + 1];
    v2f b; b.x = Bp[ka];    b.y = Bp[ka + 1];
    acc = wmma_f32(a, b, acc);
  }
  for (int r = 0; r < 8; ++r) {
    int m = r + 8 * half;
    __builtin_nontemporal_store(acc[r], &out[(size_t)(row0 + m) * M + coln]);
  }
}

// --------------------- GCN propagation (gather form) -----------------------
// out[n] = act( Ac*h[n] + Bc*( dinv[n]*sum_e dinv[src]*h[src] (+ dinv[n]^2*h[n]) ) )
__global__ __launch_bounds__(HIDN) void k_prop(
    const float* __restrict__ h, const int* __restrict__ rowptr,
    const int* __restrict__ csr, const int* __restrict__ srcp,
    const float* __restrict__ dinv, float* __restrict__ out, float Ac, float Bc,
    int self_loop, int do_relu) {
  int node = blockIdx.x, f = threadIdx.x;
  int beg = rowptr[node], end = rowptr[node + 1];
  float dn = dinv[node];
  float acc = 0.f;
  for (int e = beg; e < end; ++e) {
    int s = srcp[csr[e]];
    acc += h[(size_t)s * HIDN + f] * dinv[s];
  }
  acc *= dn;
  float hv = h[(size_t)node * HIDN + f];
  if (self_loop) acc += dn * dn * hv;
  float v = Ac * hv + Bc * acc;
  if (do_relu) v = fmaxf(v, 0.f);
  out[(size_t)node * HIDN + f] = v;
}

// ------------------- neighbor mean / std (gather form) ----------------------
__global__ __launch_bounds__(HIDN) void k_mean_var(
    const float* __restrict__ h0, const int* __restrict__ rowptr,
    const int* __restrict__ csr, const int* __restrict__ srcp,
    float* __restrict__ meann, float* __restrict__ stdn) {
  int node = blockIdx.x, f = threadIdx.x;
  int beg = rowptr[node], end = rowptr[node + 1];
  float s = 0.f;
  for (int e = beg; e < end; ++e) s += h0[(size_t)srcp[csr[e]] * HIDN + f];
  float safe = fmaxf((float)(end - beg), 1.f);
  float mu = s / safe;
  float v = 0.f;
  for (int e = beg; e < end; ++e) {
    float d = h0[(size_t)srcp[csr[e]] * HIDN + f] - mu;
    v += d * d;
  }
  v /= safe;
  meann[(size_t)node * HIDN + f] = mu;
  stdn[(size_t)node * HIDN + f] = sqrtf(v + 1e-6f);
}

// ------- per-node covariance block: gcov + mean/std/cov loss terms ---------
__global__ __launch_bounds__(256) void k_cov(
    const float* __restrict__ h0, const int* __restrict__ rowptr,
    const int* __restrict__ csr, const int* __restrict__ srcp,
    const int* __restrict__ rnd, const float* __restrict__ meann,
    const float* __restrict__ stdn, const float* __restrict__ genm,
    const float* __restrict__ gens, float* __restrict__ gcov,
    float* __restrict__ mst) {
  int node = blockIdx.x, tid = threadIdx.x;
  __shared__ float fe[KSAMP][HIDN];
  __shared__ float gm[HIDN], mn[HIDN];
  __shared__ float red[256];
  int beg = rowptr[node];
  int deg = rowptr[node + 1] - beg;
  int lim = deg < MAXNEI ? deg : MAXNEI;
  for (int i = tid; i < KSAMP * HIDN; i += 256) {
    int k = i >> 6, f = i & 63;
    float v = 0.f;
    if (deg > 0) {                       // no_nei -> zeros
      int idx = rnd[node * KSAMP + k];   // in [0,64)
      int sn = 0;                        // pad slot (-1) -> max(.,0)=node 0
      if (idx < lim) sn = srcp[csr[beg + idx]];
      v = h0[(size_t)sn * HIDN + f];
    }
    fe[k][f] = v;
  }
  if (tid < HIDN) {
    gm[tid] = genm[(size_t)node * HIDN + tid];
    mn[tid] = meann[(size_t)node * HIDN + tid];
  }
  __syncthreads();
  float csq = 0.f;
  for (int idx = tid; idx < HIDN * HIDN; idx += 256) {
    int i = idx >> 6, j = idx & 63;
    float sg = 0.f, st = 0.f;
    for (int k = 0; k < KSAMP; ++k) {
      float gi = fe[k][i] - gm[i], gj = fe[k][j] - gm[j];
      sg += gi * gj;
      float ti = fe[k][i] - mn[i], tj = fe[k][j] - mn[j];
      st += ti * tj;
    }
    float gc = sg * (1.f / (KSAMP - 1)) + (i == j ? 0.001f : 0.f);
    gcov[(size_t)node * HIDN * HIDN + idx] = gc;
    float d = (sg - st) * (1.f / (KSAMP - 1));  // reg cancels
    csq += d * d;
  }
  red[tid] = csq; __syncthreads();
  for (int s2 = 128; s2 > 0; s2 >>= 1) {
    if (tid < s2) red[tid] += red[tid + s2];
    __syncthreads();
  }
  float covt = (tid == 0) ? sqrtf(red[0]) : 0.f;
  __syncthreads();
  float mt = 0.f;
  if (tid < HIDN) {
    float d1 = gm[tid] - mn[tid];
    float d2 = gens[(size_t)node * HIDN + tid] - stdn[(size_t)node * HIDN + tid];
    mt = d1 * d1 + d2 * d2;
  }
  red[tid] = mt; __syncthreads();
  for (int s2 = 128; s2 > 0; s2 >>= 1) {
    if (tid < s2) red[tid] += red[tid + s2];
    __syncthreads();
  }
  if (tid == 0) mst[node] = red[0] + covt;
}

// ----------------------------- attention -----------------------------------

__global__ void k_att_dots(const float* __restrict__ m,
                           const float* __restrict__ a_src,
                           const float* __restrict__ a_dst, float* asn,
                           float* adn, int n) {
  int i = blockIdx.x * blockDim.x + threadIdx.x;
  if (i >= n) return;
  const float* mr = m + (size_t)i * HIDN;
  float s1 = 0.f, s2 = 0.f;
  for (int f = 0; f < HIDN; ++f) { s1 += mr[f] * a_src[f]; s2 += mr[f] * a_dst[f]; }
  asn[i] = s1; adn[i] = s2;
}

__global__ void k_att_softmax(const float* __restrict__ asn,
                              const float* __restrict__ adn,
                              const int* __restrict__ rowptr,
                              const int* __restrict__ csr,
                              const int* __restrict__ srcp, float* alpha,
                              int n) {
  int i = blockIdx.x * blockDim.x + threadIdx.x;
  if (i >= n) return;
  int beg = rowptr[i], end = rowptr[i + 1];
  if (beg == end) return;
  float ad = adn[i];
  float mx = -3.4e38f;
  for (int e = beg; e < end; ++e) {
    float v = asn[srcp[csr[e]]] + ad;
    v = (v > 0.f) ? v : 0.2f * v;        // leaky_relu 0.2
    mx = fmaxf(mx, v);
  }
  float z = 0.f;
  for (int e = beg; e < end; ++e) {
    float v = asn[srcp[csr[e]]] + ad;
    v = (v > 0.f) ? v : 0.2f * v;
    float a = expf(v - mx);
    alpha[e] = a;
    z += a;
  }
  float inv = 1.f / (z + 1e-16f);
  for (int e = beg; e < end; ++e) alpha[e] *= inv;
}

__global__ __launch_bounds__(HIDN) void k_agg_mean(
    const float* __restrict__ m, const float* __restrict__ alpha,
    const int* __restrict__ rowptr, const int* __restrict__ csr,
    const int* __restrict__ srcp, float* __restrict__ out) {
  int node = blockIdx.x, f = threadIdx.x;
  int beg = rowptr[node], end = rowptr[node + 1];
  float s = 0.f;
  for (int e = beg; e < end; ++e)
    s += alpha[e] * m[(size_t)srcp[csr[e]] * HIDN + f];
  out[(size_t)node * HIDN + f] = s;
}

// ---- cov_agg (in LDS, normalized) + h_rec = mean_agg . cov_agg ------------
__global__ __launch_bounds__(256) void k_hrec(
    const float* __restrict__ gcov, const float* __restrict__ alpha,
    const int* __restrict__ rowptr, const int* __restrict__ csr,
    const int* __restrict__ srcp, const float* __restrict__ meanagg,
    float* __restrict__ hrec) {
  int node = blockIdx.x, tid = threadIdx.x;
  __shared__ float acc[HIDN * HIDN];     // 16 KB
  __shared__ float red[256];
  __shared__ float ma[HIDN];
  __shared__ float inv_s;
  int beg = rowptr[node], end = rowptr[node + 1];
  for (int idx = tid; idx < HIDN * HIDN; idx += 256) {
    float a = 0.f;
    for (int e = beg; e < end; ++e) {
      int s = srcp[csr[e]];
      a += alpha[e] * gcov[(size_t)s * HIDN * HIDN + idx];
    }
    acc[idx] = a;
  }
  if (tid < HIDN) ma[tid] = meanagg[(size_t)node * HIDN + tid];
  __syncthreads();
  float p = 0.f;
  for (int idx = tid; idx < HIDN * HIDN; idx += 256) p += acc[idx] * acc[idx];
  red[tid] = p; __syncthreads();
  for (int s2 = 128; s2 > 0; s2 >>= 1) {
    if (tid < s2) red[tid] += red[tid + s2];
    __syncthreads();
  }
  if (tid == 0) inv_s = 1.f / (sqrtf(red[0]) + 1e-6f);
  __syncthreads();
  float inv = inv_s;
  int b = tid >> 2, q = tid & 3;
  float part = 0.f;
  for (int d = q * 16; d < q * 16 + 16; ++d) part += ma[d] * acc[b * HIDN + d];
  red[tid] = part; __syncthreads();
  if (q == 0)
    hrec[(size_t)node * HIDN + b] =
        inv * (red[tid] + red[tid + 1] + red[tid + 2] + red[tid + 3]);
}

// ----------------------------- misc elementwise -----------------------------

__global__ void k_combine(float* __restrict__ out, const float* __restrict__ a,
                          const float* __restrict__ b, float ca, float cb,
                          int n) {
  int i = blockIdx.x * blockDim.x + threadIdx.x;
  if (i < n) out[i] = ca * a[i] + cb * b[i];
}

__global__ void k_row_sqdiff(const float* __restrict__ a,
                             const float* __restrict__ b, float* out, int n) {
  int i = blockIdx.x * blockDim.x + threadIdx.x;
  if (i >= n) return;
  float s = 0.f;
  for (int f = 0; f < HIDN; ++f) {
    float d = a[(size_t)i * HIDN + f] - b[(size_t)i * HIDN + f];
    s += d * d;
  }
  out[i] = s;
}

__global__ void k_reduce_add(const float* __restrict__ v, float* out,
                             float scale, int n) {
  __shared__ float red[256];
  float s = 0.f;
  for (int i = blockIdx.x * 256 + threadIdx.x; i < n; i += gridDim.x * 256)
    s += v[i];
  red[threadIdx.x] = s; __syncthreads();
  for (int s2 = 128; s2 > 0; s2 >>= 1) {
    if (threadIdx.x < s2) red[threadIdx.x] += red[threadIdx.x + s2];
    __syncthreads();
  }
  if (threadIdx.x == 0) atomicAdd(out, red[0] * scale);
}

__global__ void k_write_loss(float* __restrict__ out,
                             const float* __restrict__ scal,
                             const float* __restrict__ h0loss, int n) {
  int i = blockIdx.x * blockDim.x + threadIdx.x;
  if (i < n) out[i] = scal[0] + h0loss[i];
}

// ---------------------------------------------------------------------------

extern "C" void kernel_launch(void* const* d_in, const int* in_sizes, int n_in,
                              void* d_out, int out_size, void* d_ws,
                              size_t ws_size, hipStream_t stream) {
  const float* x   = (const float*)d_in[0];
  const int*   ei  = (const int*)d_in[1];
  const int*   rnd = (const int*)d_in[2];
  // params in setup_inputs()/shapes insertion order
  const float* W_lin = (const float*)d_in[3];
  const float* b_lin = (const float*)d_in[4];
  const float* W_g1  = (const float*)d_in[5];
  const float* b_g1  = (const float*)d_in[6];
  const float* W_g2  = (const float*)d_in[7];
  const float* b_g2  = (const float*)d_in[8];
  const float* W_h1  = (const float*)d_in[9];
  const float* b_h1  = (const float*)d_in[10];
  const float* W_h2  = (const float*)d_in[11];
  const float* b_h2  = (const float*)d_in[12];
  const float* W_f1  = (const float*)d_in[13];
  const float* b_f1  = (const float*)d_in[14];
  const float* W_f2  = (const float*)d_in[15];
  const float* b_f2  = (const float*)d_in[16];
  const float* W_m   = (const float*)d_in[17];
  const float* b_m   = (const float*)d_in[18];
  const float* W_s   = (const float*)d_in[19];
  const float* b_s   = (const float*)d_in[20];
  const float* W_a   = (const float*)d_in[21];
  const float* a_src = (const float*)d_in[22];
  const float* a_dst = (const float*)d_in[23];
  const float* W_d1  = (const float*)d_in[24];
  const float* b_d1  = (const float*)d_in[25];
  const float* W_d2  = (const float*)d_in[26];
  const float* b_d2  = (const float*)d_in[27];
  const float* W_t1  = (const float*)d_in[28];
  const float* b_t1  = (const float*)d_in[29];
  const float* W_t2  = (const float*)d_in[30];
  const float* b_t2  = (const float*)d_in[31];

  const int N = in_sizes[0] / INDIM;     // 10000 (multiple of 16)
  const int E = in_sizes[1] / 2;         // 160000
  const int* srcp = ei;
  const int* dstp = ei + E;

  float* out   = (float*)d_out;
  float* out_x = out;                                   // [N,128]
  float* out_s = out + (size_t)N * INDIM;               // [N,N]
  float* out_l = out_s + (size_t)N * N;                 // [N]
  float* gcov  = out_s;  // N*4096 floats staged inside s_ region (written last)

  // ---- workspace layout (~63 MB) ----
  float* w = (float*)d_ws;
  auto AF = [&](size_t c) { float* p = w; w += c; return p; };
  const size_t NH = (size_t)N * HIDN;
  float* h0     = AF(NH);
  float* gpre   = AF(NH);
  float* g      = AF(NH);
  float* t2     = AF(NH);
  float* hlow   = AF(NH);
  float* z1     = AF(NH);
  float* zr     = AF(NH);
  float* z2     = AF(NH);
  float* hhigh  = AF(NH);
  float* embed  = AF(NH);
  float* f1     = AF(NH);
  float* h0rec  = AF(NH);
  float* meann  = AF(NH);
  float* stdn   = AF(NH);
  float* genm   = AF(NH);
  float* gens   = AF(NH);
  float* mfeat  = AF(NH);
  float* meanagg= AF(NH);
  float* hrec   = AF(NH);
  float* embed2 = AF(NH);
  float* d1buf  = AF(NH);
  float* t1buf  = AF(NH);
  float* hdec   = AF((size_t)N * INDIM);
  float* h0loss = AF(N);
  float* mst    = AF(N);
  float* asn    = AF(N);
  float* adn    = AF(N);
  float* dsl    = AF(N);
  float* dns    = AF(N);
  float* alpha  = AF(E);
  float* lscal  = AF(1);
  int* wi     = (int*)w;
  int* cnt    = wi;           wi += N;
  int* fill   = wi;           wi += N;
  int* rowptr = wi;           wi += N + 1;
  int* csr    = wi;           wi += E;

  const int NB = (N + 255) / 256;
  const int EB = (E + 255) / 256;
  const int MT = N / 16;                 // 625
  dim3 g64(MT), b64(128);                // NW=64 GEMMs: 4 waves
  dim3 b128(256);                        // NW=128 GEMMs: 8 waves

  // ---- CSR ----
  k_zero_i32<<<NB, 256, 0, stream>>>(cnt, N);
  k_zero_i32<<<NB, 256, 0, stream>>>(fill, N);
  k_zero_f32<<<1, 32, 0, stream>>>(lscal, 1);
  k_count_deg<<<EB, 256, 0, stream>>>(dstp, cnt, E);
  k_prefix<<<1, 32, 0, stream>>>(cnt, rowptr, N);
  k_fill_csr<<<EB, 256, 0, stream>>>(dstp, rowptr, fill, csr, E);
  k_sort_csr<<<NB, 256, 0, stream>>>(rowptr, csr, N);
  k_dinv<<<NB, 256, 0, stream>>>(cnt, dsl, dns, N);

  // ---- encoders ----
  k_gemm<INDIM, HIDN, 0><<<g64, b64, 0, stream>>>(x, W_lin, b_lin, h0, N);
  k_gemm<INDIM, HIDN, 0><<<g64, b64, 0, stream>>>(x, W_h1, b_h1, z1, N);
  k_gemm<HIDN, HIDN, 0><<<g64, b64, 0, stream>>>(h0, W_g1, b_g1, gpre, N);
  k_prop<<<N, HIDN, 0, stream>>>(gpre, rowptr, csr, srcp, dsl, g, 0.f, 1.f, 1, 1);
  k_gemm<HIDN, HIDN, 0><<<g64, b64, 0, stream>>>(g, W_g2, b_g2, t2, N);
  k_prop<<<N, HIDN, 0, stream>>>(t2, rowptr, csr, srcp, dsl, hlow, 0.f, 1.f, 1, 0);
  k_prop<<<N, HIDN, 0, stream>>>(z1, rowptr, csr, srcp, dns, zr, 1.f, -1.f, 0, 1);
  k_gemm<HIDN, HIDN, 0><<<g64, b64, 0, stream>>>(zr, W_h2, b_h2, z2, N);
  k_prop<<<N, HIDN, 0, stream>>>(z2, rowptr, csr, srcp, dns, hhigh, 1.f, -1.f, 0, 0);
  k_combine<<<(int)((NH + 255) / 256), 256, 0, stream>>>(embed, hlow, hhigh,
                                                         0.5f, 0.5f, (int)NH);

  // ---- feature decoder + loss ----
  k_gemm<HIDN, HIDN, 1><<<g64, b64, 0, stream>>>(embed, W_f1, b_f1, f1, N);
  k_gemm<HIDN, HIDN, 0><<<g64, b64, 0, stream>>>(f1, W_f2, b_f2, h0rec, N);
  k_row_sqdiff<<<NB, 256, 0, stream>>>(h0rec, h0, h0loss, N);

  // ---- neighborhood stats + covariance ----
  k_mean_var<<<N, HIDN, 0, stream>>>(h0, rowptr, csr, srcp, meann, stdn);
  k_gemm<HIDN, HIDN, 0><<<g64, b64, 0, stream>>>(hlow, W_m, b_m, genm, N);
  k_gemm<HIDN, HIDN, 3><<<g64, b64, 0, stream>>>(hhigh, W_s, b_s, gens, N);
  k_cov<<<N, 256, 0, stream>>>(h0, rowptr, csr, srcp, rnd, meann, stdn, genm,
                               gens, gcov, mst);
  k_reduce_add<<<64, 256, 0, stream>>>(mst, lscal, 1.f / (float)N, N);

  // ---- GAT aggregation ----
  k_gemm<HIDN, HIDN, 0><<<g64, b64, 0, stream>>>(meann, W_a, nullptr, mfeat, N);
  k_att_dots<<<NB, 256, 0, stream>>>(mfeat, a_src, a_dst, asn, adn, N);
  k_att_softmax<<<NB, 256, 0, stream>>>(asn, adn, rowptr, csr, srcp, alpha, N);
  k_agg_mean<<<N, HIDN, 0, stream>>>(mfeat, alpha, rowptr, csr, srcp, meanagg);
  k_hrec<<<N, 256, 0, stream>>>(gcov, alpha, rowptr, csr, srcp, meanagg, hrec);
  k_combine<<<(int)((NH + 255) / 256), 256, 0, stream>>>(embed2, embed, hrec,
                                                         0.8f, 0.2f, (int)NH);

  // ---- decoders ----
  k_gemm<HIDN, HIDN, 2><<<g64, b64, 0, stream>>>(embed2, W_d1, b_d1, d1buf, N);
  k_gemm<HIDN, INDIM, 0><<<g64, b128, 0, stream>>>(d1buf, W_d2, b_d2, out_x, N);
  k_gemm<HIDN, HIDN, 2><<<g64, b64, 0, stream>>>(embed2, W_t1, b_t1, t1buf, N);
  k_gemm<HIDN, INDIM, 0><<<g64, b128, 0, stream>>>(t1buf, W_t2, b_t2, hdec, N);

  // ---- s_ = h_ h_^T (overwrites gcov staging region; must be last) ----
  dim3 gs(MT, (MT + 3) / 4);
  k_gemm_sim<<<gs, 128, 0, stream>>>(hdec, out_s, N);

  // ---- loss vector ----
  k_write_loss<<<NB, 256, 0, stream>>>(out_l, lscal, h0loss, N);
}